// Adder8Bit_23407571764128
// MI455X (gfx1250) — compile-verified
//
#include <hip/hip_runtime.h>
#include <hip/hip_bf16.h>

// 8-bit ripple-carry adder on {0,1} floats.
// Memory-bound streamer (~832MB @ 23.3TB/s -> ~36us floor): CDNA5 async
// global->LDS DMA double-buffer pipeline, ds_load_b128 consume, NT stores.

typedef float v4f __attribute__((ext_vector_type(4)));
typedef int   v4i __attribute__((ext_vector_type(4)));

#define TPB 256

#if defined(__HIP_DEVICE_COMPILE__) && __has_builtin(__builtin_amdgcn_global_load_async_to_lds_b128)
  #define USE_ASYNC 1
#else
  #define USE_ASYNC 0
  #if defined(__HIP_DEVICE_COMPILE__)
    #warning "gfx1250 async-to-LDS builtin NOT available: using direct-load fallback path"
  #endif
#endif

#if USE_ASYNC
  #define GAS __attribute__((address_space(1)))
  #define LAS __attribute__((address_space(3)))
  // Builtin signature: (int4 AS1*, int4 AS3*, imm off, imm cpol).
  // INST_OFFSET is added to BOTH the global and the LDS address (CDNA5 ISA),
  // so one base pointer pair + imm offsets 0/16 covers a 32B row.
  #define ASYNC_B128(gsrc, ldst, off) \
      __builtin_amdgcn_global_load_async_to_lds_b128( \
          (GAS v4i*)(GAS void*)(void*)(gsrc), \
          (LAS v4i*)(LAS void*)(void*)(ldst), (off), 0)

  #if __has_builtin(__builtin_amdgcn_s_wait_asynccnt)
    #define WAIT_ASYNC(n) do { __builtin_amdgcn_s_wait_asynccnt(n); \
                               asm volatile("" ::: "memory"); } while (0)
  #else
    #define WAIT_ASYNC(n) asm volatile("s_wait_asynccnt %0" :: "i"(n) : "memory")
  #endif
#endif

// Arithmetic full adder on {0,1} floats:
//   x  = a ^ b    = a + b - 2ab
//   s  = x ^ c    = x + c - 2xc
//   c' = ab + c*x
__device__ __forceinline__ void ripple8(const float a[8], const float b[8],
                                        float cin, float s[8], float& cout) {
    float c = cin;
#pragma unroll
    for (int i = 7; i >= 0; --i) {          // LSB (bit 7) -> MSB (bit 0)
        float ab = a[i] * b[i];
        float x  = fmaf(-2.0f, ab, a[i] + b[i]);
        float xc = x * c;
        s[i]     = fmaf(-2.0f, xc, x + c);
        c        = ab + xc;
    }
    cout = c;
}

__device__ __forceinline__ void compute_store(v4f a0, v4f a1, v4f b0, v4f b1,
                                              float cin, float* __restrict__ out,
                                              long N, long r) {
    const float a[8] = { a0.x, a0.y, a0.z, a0.w, a1.x, a1.y, a1.z, a1.w };
    const float b[8] = { b0.x, b0.y, b0.z, b0.w, b1.x, b1.y, b1.z, b1.w };
    float s[8], c;
    ripple8(a, b, cin, s, c);
    v4f o0 = { s[0], s[1], s[2], s[3] };
    v4f o1 = { s[4], s[5], s[6], s[7] };
    v4f* orow = (v4f*)(out + r * 8);
    __builtin_nontemporal_store(o0, orow);
    __builtin_nontemporal_store(o1, orow + 1);
    __builtin_nontemporal_store(c, out + N * 8 + r);
}

__global__ __launch_bounds__(TPB) void adder8_kernel(
        const float* __restrict__ A, const float* __restrict__ B,
        const float* __restrict__ Cin, float* __restrict__ out, long N) {
    const int  tid    = (int)threadIdx.x;
    const long stride = (long)gridDim.x * TPB;
    long       r      = (long)blockIdx.x * TPB + tid;

#if USE_ASYNC
    // 2 buffers x 256 threads x 2 float4 x (A,B) = 32KB LDS / block.
    // Typed as v4f and indexed directly so consume reads stay addrspace(3)
    // (ds_load_b128), not flat.
    __shared__ __align__(16) v4f bufA[2][TPB * 2];
    __shared__ __align__(16) v4f bufB[2][TPB * 2];
    const int slot = tid * 2;

    // Prologue: fill buffer 0 (4 async B128 copies = one A row + one B row).
    if (r < N) {
        ASYNC_B128(A + r * 8, &bufA[0][slot], 0);
        ASYNC_B128(A + r * 8, &bufA[0][slot], 16);
        ASYNC_B128(B + r * 8, &bufB[0][slot], 0);
        ASYNC_B128(B + r * 8, &bufB[0][slot], 16);
    }
    int cur = 0;
    while (r < N) {
        const long rn = r + stride;
        const bool more = rn < N;
        if (more) {  // prefetch next chunk into the other buffer
            const int nxt = cur ^ 1;
            ASYNC_B128(A + rn * 8, &bufA[nxt][slot], 0);
            ASYNC_B128(A + rn * 8, &bufA[nxt][slot], 16);
            ASYNC_B128(B + rn * 8, &bufB[nxt][slot], 0);
            ASYNC_B128(B + rn * 8, &bufB[nxt][slot], 16);
        }
        const float cin = __builtin_nontemporal_load(Cin + r);
        // Async loads complete in order: leaving <=4 outstanding means the
        // current buffer's 4 copies have landed in LDS.
        if (more) WAIT_ASYNC(4); else WAIT_ASYNC(0);

        v4f a0 = bufA[cur][slot], a1 = bufA[cur][slot + 1];
        v4f b0 = bufB[cur][slot], b1 = bufB[cur][slot + 1];
        compute_store(a0, a1, b0, b1, cin, out, N, r);

        r = rn;
        cur ^= 1;
    }
#else
    for (; r < N; r += stride) {
        const v4f* va = (const v4f*)(A + r * 8);
        const v4f* vb = (const v4f*)(B + r * 8);
        v4f a0 = __builtin_nontemporal_load(va);
        v4f a1 = __builtin_nontemporal_load(va + 1);
        v4f b0 = __builtin_nontemporal_load(vb);
        v4f b1 = __builtin_nontemporal_load(vb + 1);
        const float cin = __builtin_nontemporal_load(Cin + r);
        compute_store(a0, a1, b0, b1, cin, out, N, r);
    }
#endif
}

extern "C" void kernel_launch(void* const* d_in, const int* in_sizes, int n_in,
                              void* d_out, int out_size, void* d_ws, size_t ws_size,
                              hipStream_t stream) {
    (void)n_in; (void)out_size; (void)d_ws; (void)ws_size;
    const float* A   = (const float*)d_in[0];
    const float* B   = (const float*)d_in[1];
    const float* Cin = (const float*)d_in[2];
    float*       out = (float*)d_out;

    const long N = (long)in_sizes[0] / 8;   // rows
    long nblk = (N + TPB - 1) / TPB;
    int blocks = (int)(nblk < 4096 ? nblk : 4096);  // grid-stride: 8 chunks/thread at N=2^23
    if (blocks < 1) blocks = 1;

    adder8_kernel<<<blocks, TPB, 0, stream>>>(A, B, Cin, out, N);
}